// IndRNN_56951266345464
// MI455X (gfx1250) — compile-verified
//
#include <hip/hip_runtime.h>

typedef __attribute__((ext_vector_type(2))) float v2f;
typedef __attribute__((ext_vector_type(8))) float v8f;

#define SEQ   1024
#define BATCH 64
#define IN    512
#define HID   2048
#define BH    (BATCH * HID)   // 131072 elements per timestep

#define BM 128
#define BN 128
#define KC 32
#define LDSS 36   // padded LDS row stride (floats): 36*r mod 64 distinct for 16 rows

// ---------------------------------------------------------------------------
// Tiled fp32 WMMA GEMM:  out[M, 2048] = A[M, K] * W[2048, K]^T + bias
// (B-matrix element B[k][n] == W[n][k], so LDS keeps W in native [n][k] order)
// ---------------------------------------------------------------------------
__global__ __launch_bounds__(256) void indrnn_gemm(const float* __restrict__ A,
                                                   const float* __restrict__ W,
                                                   const float* __restrict__ bias,
                                                   float* __restrict__ out,
                                                   int M, int K)
{
    __shared__ float ldsA[BM][LDSS];
    __shared__ float ldsB[BN][LDSS];

    const int tid  = threadIdx.x;
    const int lane = tid & 31;
    const int wid  = tid >> 5;      // 0..7
    const int wm   = wid >> 1;      // 0..3 : 32-row strip within block
    const int wn   = wid & 1;       // 0..1 : 64-col strip within block
    const int Lm   = lane & 15;
    const int hi   = lane >> 4;     // 0 or 1 (half-wave)

    const long rowBlock = (long)blockIdx.y * BM;
    const long colBlock = (long)blockIdx.x * BN;

    v8f acc[2][4];
#pragma unroll
    for (int i = 0; i < 2; ++i)
#pragma unroll
        for (int j = 0; j < 4; ++j)
            acc[i][j] = (v8f){0.f, 0.f, 0.f, 0.f, 0.f, 0.f, 0.f, 0.f};

    // Staging: each of 256 threads copies 16 floats of A and 16 floats of W
    const int ldRow = tid >> 1;        // 0..127
    const int ldK   = (tid & 1) * 16;  // 0 or 16

    const float* Ag = A + (rowBlock + ldRow) * (long)K + ldK;
    const float* Wg = W + (colBlock + ldRow) * (long)K + ldK;

    for (int k0 = 0; k0 < K; k0 += KC) {
        float4 a0 = *(const float4*)(Ag + k0);
        float4 a1 = *(const float4*)(Ag + k0 + 4);
        float4 a2 = *(const float4*)(Ag + k0 + 8);
        float4 a3 = *(const float4*)(Ag + k0 + 12);
        float4 w0 = *(const float4*)(Wg + k0);
        float4 w1 = *(const float4*)(Wg + k0 + 4);
        float4 w2 = *(const float4*)(Wg + k0 + 8);
        float4 w3 = *(const float4*)(Wg + k0 + 12);

        __syncthreads();   // previous stage fully consumed
        *(float4*)&ldsA[ldRow][ldK +  0] = a0;
        *(float4*)&ldsA[ldRow][ldK +  4] = a1;
        *(float4*)&ldsA[ldRow][ldK +  8] = a2;
        *(float4*)&ldsA[ldRow][ldK + 12] = a3;
        *(float4*)&ldsB[ldRow][ldK +  0] = w0;
        *(float4*)&ldsB[ldRow][ldK +  4] = w1;
        *(float4*)&ldsB[ldRow][ldK +  8] = w2;
        *(float4*)&ldsB[ldRow][ldK + 12] = w3;
        __syncthreads();   // stage visible

#pragma unroll
        for (int kc = 0; kc < KC; kc += 4) {
            const int kk = kc + 2 * hi;    // half-wave K split per A/B VGPR layout
            v2f af[2], bf[4];
#pragma unroll
            for (int ms = 0; ms < 2; ++ms) {
                const int r = 32 * wm + 16 * ms + Lm;
                af[ms].x = ldsA[r][kk];
                af[ms].y = ldsA[r][kk + 1];
            }
#pragma unroll
            for (int ns = 0; ns < 4; ++ns) {
                const int c = 64 * wn + 16 * ns + Lm;
                bf[ns].x = ldsB[c][kk];
                bf[ns].y = ldsB[c][kk + 1];
            }
#pragma unroll
            for (int ms = 0; ms < 2; ++ms)
#pragma unroll
                for (int ns = 0; ns < 4; ++ns)
                    acc[ms][ns] = __builtin_amdgcn_wmma_f32_16x16x4_f32(
                        false, af[ms], false, bf[ns],
                        (short)0, acc[ms][ns], false, false);
        }
    }

    // Epilogue: D element (VGPR r, lane) -> row r + 8*hi, col Lm (16x16 f32 layout)
#pragma unroll
    for (int ms = 0; ms < 2; ++ms) {
        const long mrow = rowBlock + 32 * wm + 16 * ms + 8 * hi;
#pragma unroll
        for (int ns = 0; ns < 4; ++ns) {
            const long ncol = colBlock + 64 * wn + 16 * ns + Lm;
            const float bv = bias[ncol];
#pragma unroll
            for (int r = 0; r < 8; ++r)
                out[(mrow + r) * (long)HID + ncol] = acc[ms][ns][r] + bv;
        }
    }
}

// ---------------------------------------------------------------------------
// Diagonal recurrence: h[t] = relu(pre[t] + w_hh * h[t-1]), one thread/(b,h).
// Safe to run in-place (pre == out). Carry holds h across time chunks.
// ---------------------------------------------------------------------------
__global__ __launch_bounds__(256) void indrnn_scan(const float* __restrict__ pre,
                                                   float* __restrict__ out,
                                                   const float* __restrict__ w_hh,
                                                   float* __restrict__ carry,
                                                   int Tc)
{
    const int idx = blockIdx.x * blockDim.x + threadIdx.x;   // 0..BH-1
    const float w = w_hh[idx & (HID - 1)];
    float h = carry[idx];
    const float* p = pre + idx;
    float* o = out + idx;
    for (int t = 0; t < Tc; ++t) {
        float v = p[(size_t)t * BH] + w * h;
        h = v > 0.0f ? v : 0.0f;
        o[(size_t)t * BH] = h;
    }
    carry[idx] = h;
}

__global__ void indrnn_zero(float* __restrict__ p, int n)
{
    int i = blockIdx.x * blockDim.x + threadIdx.x;
    if (i < n) p[i] = 0.0f;
}

// ---------------------------------------------------------------------------
extern "C" void kernel_launch(void* const* d_in, const int* in_sizes, int n_in,
                              void* d_out, int out_size, void* d_ws, size_t ws_size,
                              hipStream_t stream)
{
    (void)in_sizes; (void)n_in; (void)out_size;

    const float* x     = (const float*)d_in[0];
    const float* w_ih0 = (const float*)d_in[1];
    const float* w_hh0 = (const float*)d_in[2];
    const float* b0    = (const float*)d_in[3];
    const float* w_ih1 = (const float*)d_in[4];
    const float* w_hh1 = (const float*)d_in[5];
    const float* b1    = (const float*)d_in[6];
    float* out = (float*)d_out;

    float* carry0 = (float*)d_ws;            // [BH] layer-1 hidden carry
    float* carry1 = carry0 + BH;             // [BH] layer-2 hidden carry
    float* buf    = carry1 + BH;             // [Tc, BATCH, HID] chunk buffer

    // Largest time chunk whose pre1/h1 buffer fits in the remaining workspace.
    const size_t carryBytes = (size_t)2 * BH * sizeof(float);
    const size_t avail = ws_size > carryBytes ? ws_size - carryBytes : 0;
    int Tc = 2;
    const int cands[10] = {1024, 512, 256, 128, 64, 32, 16, 8, 4, 2};
    for (int i = 0; i < 10; ++i) {
        if ((size_t)cands[i] * BH * sizeof(float) <= avail) { Tc = cands[i]; break; }
    }

    indrnn_zero<<<(2 * BH + 255) / 256, 256, 0, stream>>>(carry0, 2 * BH);

    for (int t0 = 0; t0 < SEQ; t0 += Tc) {
        const int Mc = Tc * BATCH;                    // rows this chunk (multiple of 128)
        const float* xc = x + (size_t)t0 * BATCH * IN;
        float* oc = out + (size_t)t0 * BH;

        dim3 grid(HID / BN, Mc / BM);                 // N-major: A rows stay L2-hot
        // Layer 1: input projection + bias -> buf, then in-place scan
        indrnn_gemm<<<grid, 256, 0, stream>>>(xc, w_ih0, b0, buf, Mc, IN);
        indrnn_scan<<<BH / 256, 256, 0, stream>>>(buf, buf, w_hh0, carry0, Tc);
        // Layer 2: projection of h1 straight into d_out, then in-place scan
        indrnn_gemm<<<grid, 256, 0, stream>>>(buf, w_ih1, b1, oc, Mc, HID);
        indrnn_scan<<<BH / 256, 256, 0, stream>>>(oc, oc, w_hh1, carry1, Tc);
    }
}